// NoisyTopKRouter_20066087206922
// MI455X (gfx1250) — compile-verified
//
#include <hip/hip_runtime.h>
#include <math.h>

typedef float v2f __attribute__((ext_vector_type(2)));
typedef float v4f __attribute__((ext_vector_type(4)));
typedef float v8f __attribute__((ext_vector_type(8)));

#define HDIM 2048
#define EXP 16
#define TOK 16384           // B*S = 4*4096
#define TILES (TOK / 16)    // 1024 tiles of 16 tokens
#define WAVES_PER_BLOCK 4
#define STEPS (HDIM / 4)    // 512 K-steps of 4

// ws layout (floats): [0:16) usage, [16:32) imp, [32] ent, [33] z, [64:...) packed B (256KB)
#define WPK_OFF 64

// ---------------------------------------------------------------- zero accumulators
__global__ void rtr_zero_ws(float* ws) {
    if (threadIdx.x < 64) ws[threadIdx.x] = 0.0f;
}

// ---------------------------------------------------------------- pack gate+noise weights
// into WMMA B lane layout: entry i = s*32 + half*16 + e  ->  16B {g0,g1,n0,n1}
// where gK = gw[e][4s + 2*half + K], nK = nw[e][4s + 2*half + K]
__global__ __launch_bounds__(256) void rtr_pack_w(
    const float* __restrict__ gw, const float* __restrict__ nw,
    float* __restrict__ wpk)
{
    const int i    = blockIdx.x * 256 + threadIdx.x;   // 0 .. 16383
    const int e    = i & 15;
    const int half = (i >> 4) & 1;
    const int s    = i >> 5;
    const size_t src = (size_t)e * HDIM + s * 4 + half * 2;
    v2f g = *(const v2f*)(gw + src);
    v2f n = *(const v2f*)(nw + src);
    v4f p = { g.x, g.y, n.x, n.y };
    *(v4f*)(wpk + (size_t)i * 4) = p;
}

// ---------------------------------------------------------------- main fused kernel
__global__ __launch_bounds__(128) void rtr_main(
    const float* __restrict__ hs,     // [TOK, HDIM]
    const float* __restrict__ nb,     // [EXP]
    const float* __restrict__ noise,  // [TOK, EXP]
    float* __restrict__ out,          // dispatch | combine | aux
    float* __restrict__ ws)           // accumulators + packed weights
{
    __shared__ float s_usage[EXP];
    __shared__ float s_imp[EXP];
    __shared__ float s_ent;
    __shared__ float s_z;

    const int tid = threadIdx.x;
    if (tid < EXP) { s_usage[tid] = 0.0f; s_imp[tid] = 0.0f; }
    if (tid == 0)  { s_ent = 0.0f; s_z = 0.0f; }
    __syncthreads();

    const int lane  = tid & 31;
    const int wave  = tid >> 5;
    const int tile  = blockIdx.x * WAVES_PER_BLOCK + wave;
    const int half  = lane >> 4;      // 0: lanes 0-15, 1: lanes 16-31
    const int e     = lane & 15;      // expert (B/C column), token row for A
    const int klo   = half * 2;       // K offset pair within 4-wide step
    const int tbase = tile * 16;

    // A: hs row = token (lane&15); per 4-K step lane supplies K = klo, klo+1
    const float* ap = hs + (size_t)(tbase + e) * HDIM + klo;
    // packed B: lane reads entry (half*16 + e), stride 32 entries (128 floats) per step
    const float* bp = ws + WPK_OFF + (size_t)(half * 16 + e) * 4;

    v8f cg = {0.f,0.f,0.f,0.f,0.f,0.f,0.f,0.f};
    v8f cn = {0.f,0.f,0.f,0.f,0.f,0.f,0.f,0.f};

    for (int h = 0; h < HDIM; h += 32) {          // 8 K-steps per iteration
        __builtin_prefetch(ap + h + 2048, 0, 1);  // stream hs ~8KB ahead
        #pragma unroll
        for (int u = 0; u < 8; ++u) {
            const int hh = h + u * 4;
            v2f a = *(const v2f*)(ap + hh);
            v4f b = *(const v4f*)(bp + (size_t)(hh >> 2) * 128);
            v2f bg = __builtin_shufflevector(b, b, 0, 1);
            v2f bn = __builtin_shufflevector(b, b, 2, 3);
            cg = __builtin_amdgcn_wmma_f32_16x16x4_f32(false, a, false, bg, (short)0, cg, false, false);
            cn = __builtin_amdgcn_wmma_f32_16x16x4_f32(false, a, false, bn, (short)0, cn, false, false);
        }
    }

    // ---- epilogue: accum entry j / this lane = logits[token tbase+j+8*half][expert e]
    const float nbv = nb[e];
    float* outC = out + (size_t)TOK * EXP * 2;    // combine_weights base

    float l_usage = 0.0f, l_imp = 0.0f, l_ent = 0.0f, l_z = 0.0f;

    #pragma unroll
    for (int j = 0; j < 8; ++j) {
        const int t = tbase + j + half * 8;
        float gl = cg[j];
        float nl = cn[j] + nbv;
        // stable softplus
        float sp = (nl > 0.0f) ? (nl + log1pf(expf(-nl))) : log1pf(expf(nl));
        float x  = gl + noise[(size_t)t * EXP + e] * sp;   // NOISE_STD = 1.0

        // softmax across the 16 lanes of this half
        float mv = x;
        #pragma unroll
        for (int mk = 1; mk < 16; mk <<= 1) mv = fmaxf(mv, __shfl_xor(mv, mk, 32));
        float p = expf(x - mv);
        float ss = p;
        #pragma unroll
        for (int mk = 1; mk < 16; mk <<= 1) ss += __shfl_xor(ss, mk, 32);
        float score = p / ss;

        // top-1 (value, index) with lowest-index tie-break
        float v1 = score; int i1 = e;
        #pragma unroll
        for (int mk = 1; mk < 16; mk <<= 1) {
            float ov = __shfl_xor(v1, mk, 32);
            int   oi = __shfl_xor(i1, mk, 32);
            if (ov > v1 || (ov == v1 && oi < i1)) { v1 = ov; i1 = oi; }
        }
        // top-2
        float v2 = (e == i1) ? -1e30f : score; int i2 = e;
        #pragma unroll
        for (int mk = 1; mk < 16; mk <<= 1) {
            float ov = __shfl_xor(v2, mk, 32);
            int   oi = __shfl_xor(i2, mk, 32);
            if (ov > v2 || (ov == v2 && oi < i2)) { v2 = ov; i2 = oi; }
        }
        float denom = v1 + v2;
        float w1 = v1 / denom, w2 = v2 / denom;

        float dm0 = (e == i1) ? 1.0f : 0.0f;
        float dm1 = (e == i2) ? 1.0f : 0.0f;
        size_t off = ((size_t)t * EXP + e) * 2;
        v2f dmv = { dm0, dm1 };
        v2f cwv = { dm0 * w1, dm1 * w2 };
        *(v2f*)(out  + off) = dmv;
        *(v2f*)(outC + off) = cwv;

        l_usage += dm0 + dm1;
        l_imp   += score;
        l_ent   += score * logf(score + 1e-10f);
        // z-loss: logsumexp over scores (per reference, on scores not logits)
        float q = expf(score);
        float zs = q;
        #pragma unroll
        for (int mk = 1; mk < 16; mk <<= 1) zs += __shfl_xor(zs, mk, 32);
        if (e == 0) { float lz = logf(zs); l_z += lz * lz; }
    }

    // wave-level reduction for scalar sums (l_z nonzero only on lanes 0,16)
    #pragma unroll
    for (int mk = 16; mk >= 1; mk >>= 1) {
        l_ent += __shfl_xor(l_ent, mk, 32);
        l_z   += __shfl_xor(l_z,   mk, 32);
    }
    atomicAdd(&s_usage[e], l_usage);
    atomicAdd(&s_imp[e],   l_imp);
    if (lane == 0) { atomicAdd(&s_ent, l_ent); atomicAdd(&s_z, l_z); }
    __syncthreads();

    if (tid < EXP) {
        atomicAdd(&ws[tid],       s_usage[tid]);
        atomicAdd(&ws[EXP + tid], s_imp[tid]);
    }
    if (tid == 0) {
        atomicAdd(&ws[32], s_ent);
        atomicAdd(&ws[33], s_z);
    }
}

// ---------------------------------------------------------------- finalize aux losses
__global__ void rtr_finalize(const float* __restrict__ ws, float* __restrict__ out) {
    if (threadIdx.x != 0 || blockIdx.x != 0) return;
    const float Tf = (float)TOK;
    float tot = 0.0f;
    for (int e = 0; e < EXP; ++e) tot += ws[e];

    float eu[EXP];
    const float uni = 1.0f / (float)EXP;
    float load_loss = 0.0f;
    for (int e = 0; e < EXP; ++e) {
        eu[e] = ws[e] / (tot + 1e-10f);
        load_loss += uni * (logf(uni) - logf(eu[e] + 1e-10f));
    }
    load_loss /= (float)EXP;

    float im[EXP], msum = 0.0f;
    for (int e = 0; e < EXP; ++e) { im[e] = ws[EXP + e] / Tf; msum += im[e]; }
    float mean = msum / (float)EXP;
    float var = 0.0f;
    for (int e = 0; e < EXP; ++e) { float d = im[e] - mean; var += d * d; }
    var /= (float)EXP;
    float importance_loss = var / (mean * mean + 1e-10f);

    float routing_entropy_scores = -ws[32] / Tf;
    float entropy_reg = (logf((float)EXP) - routing_entropy_scores) * 10.0f;
    float z_loss = ws[33] / Tf;

    float re_usage = 0.0f;
    for (int e = 0; e < EXP; ++e) re_usage -= eu[e] * logf(eu[e] + 1e-10f);

    float* aux = out + (size_t)2 * TOK * EXP * 2;
    aux[0] = load_loss;
    aux[1] = importance_loss;
    aux[2] = z_loss;
    aux[3] = entropy_reg;
    for (int e = 0; e < EXP; ++e) aux[4 + e] = eu[e];
    aux[20] = re_usage;
}

// ---------------------------------------------------------------- launch
extern "C" void kernel_launch(void* const* d_in, const int* in_sizes, int n_in,
                              void* d_out, int out_size, void* d_ws, size_t ws_size,
                              hipStream_t stream) {
    const float* hs    = (const float*)d_in[0];
    const float* gw    = (const float*)d_in[1];
    const float* nw    = (const float*)d_in[2];
    const float* nb    = (const float*)d_in[3];
    const float* noise = (const float*)d_in[4];
    float* out = (float*)d_out;
    float* ws  = (float*)d_ws;

    rtr_zero_ws<<<1, 64, 0, stream>>>(ws);
    rtr_pack_w<<<(STEPS * 32) / 256, 256, 0, stream>>>(gw, nw, ws + WPK_OFF);
    rtr_main<<<TILES / WAVES_PER_BLOCK, 32 * WAVES_PER_BLOCK, 0, stream>>>(
        hs, nb, noise, out, ws);
    rtr_finalize<<<1, 32, 0, stream>>>(ws, out);
}